// attention_H_18107582120775
// MI455X (gfx1250) — compile-verified
//
#include <hip/hip_runtime.h>
#include <math.h>

// ---------------------------------------------------------------------------
// GCN 3-layer forward for MI455X (gfx1250, wave32).
//   layer: h' = scatter_add_{e:(s,d)} norm_e * (h@W)[s] + dis[i]^2*(h@W)[i] + b
//   norm_e = rsqrt(deg[s]) * rsqrt(deg[d]);  deg = in-degree incl. self loop.
// Dense transforms: V_WMMA_F32_16X16X4_F32 (exact fp32, tiny FLOP count).
// Aggregation: packed 16B edge records + float4 gathers + f32 global atomics;
// all hot arrays (<=78 MB) are L2-resident on the 192 MB L2.
// ---------------------------------------------------------------------------

typedef float v2f __attribute__((ext_vector_type(2)));
typedef float v8f __attribute__((ext_vector_type(8)));

static inline long long cdiv_ll(long long a, long long b) { return (a + b - 1) / b; }

__global__ void fill_kernel(float* __restrict__ p, float v, unsigned n) {
  unsigned i = blockIdx.x * blockDim.x + threadIdx.x;
  if (i < n) p[i] = v;
}

// deg[dst] += 1 per edge (deg pre-filled with 1.0 for the self loop)
__global__ void deg_kernel(const int* __restrict__ ei, float* __restrict__ deg, int E) {
  int e = blockIdx.x * blockDim.x + threadIdx.x;
  if (e < E) atomicAdd(&deg[ei[E + e]], 1.0f);
}

__global__ void dis_kernel(const float* __restrict__ deg, float* __restrict__ dis, int N) {
  int i = blockIdx.x * blockDim.x + threadIdx.x;
  if (i < N) { float d = deg[i]; dis[i] = d > 0.0f ? rsqrtf(d) : 0.0f; }
}

// Pack {src, dst, norm} into one 16B record per edge (single b128 per worker
// thread later, broadcast-friendly since an edge's workers share the address).
__global__ void edge_kernel(const int* __restrict__ ei, const float* __restrict__ dis,
                            int4* __restrict__ edges, int E) {
  int e = blockIdx.x * blockDim.x + threadIdx.x;
  if (e >= E) return;
  int s = ei[e], d = ei[E + e];
  int4 r;
  r.x = s;
  r.y = d;
  r.z = __float_as_int(dis[s] * dis[d]);
  r.w = 0;
  edges[e] = r;
}

// out[N,OUT] = H[N,K] @ W[K,OUT]   (K mult of 4, OUT mult of 16)
// One wave per 16x16 output tile; K-loop in steps of 4 with
// V_WMMA_F32_16X16X4_F32.
//  A 16x4 f32 layout: lanes 0-15 row m, VGPR0/1 = K 0/1; lanes 16-31 = K 2/3.
//  B 4x16 f32 layout: lane&15 = column n, lane-half selects K rows {0,1}/{2,3}.
//  C/D 16x16 f32:     VGPR r holds row (r + 8*(lane>>4)), col = lane&15.
__global__ void gemm_wmma_f32(const float* __restrict__ H, const float* __restrict__ W,
                              float* __restrict__ out, int N, int K, int OUT,
                              int nColTiles, int nTiles) {
  int wave = (int)(((unsigned)blockIdx.x * blockDim.x + threadIdx.x) >> 5);
  if (wave >= nTiles) return;                 // wave-uniform; EXEC stays all-ones
  int lane = threadIdx.x & 31;
  int rt = wave / nColTiles, ct = wave % nColTiles;
  int half = lane >> 4;                       // 0: K 0,1  | 1: K 2,3
  int m = lane & 15;
  int row = rt * 16 + m;
  int rowc = row < N ? row : N - 1;           // clamp loads for ragged tail
  int col = ct * 16 + m;
  const float* hrow = H + (unsigned)rowc * (unsigned)K;

  v8f acc = {};
  for (int kk = 0; kk < K; kk += 4) {
    int ka = kk + half * 2;
    v2f a = *(const v2f*)(hrow + ka);         // 8B packed A load
    v2f b;
    b.x = W[(unsigned)ka * (unsigned)OUT + col];
    b.y = W[(unsigned)(ka + 1) * (unsigned)OUT + col];
    acc = __builtin_amdgcn_wmma_f32_16x16x4_f32(
        /*neg_a=*/false, a, /*neg_b=*/false, b,
        /*c_mod=*/(short)0, acc, /*reuse_a=*/false, /*reuse_b=*/false);
  }
#pragma unroll
  for (int r = 0; r < 8; ++r) {
    int mrow = rt * 16 + r + 8 * half;
    if (mrow < N) out[(unsigned)mrow * (unsigned)OUT + col] = acc[r];
  }
}

// agg[dst,:] += ht[src,:] * norm ;  C = 1<<LOGC channels, C/4 lanes per edge.
template <int LOGC>
__global__ void scatter_kernel(const float* __restrict__ ht, const int4* __restrict__ edges,
                               float* __restrict__ agg, unsigned E) {
  unsigned t = blockIdx.x * blockDim.x + threadIdx.x;
  unsigned e = t >> (LOGC - 2);
  if (e >= E) return;
  unsigned c4 = (t & ((1u << (LOGC - 2)) - 1u)) << 2;
  const int4 r = edges[e];                      // one b128, cache-broadcast
  unsigned src = (unsigned)r.x, dst = (unsigned)r.y;
  float nrm = __int_as_float(r.z);
  const float4 v = *(const float4*)(ht + (src << LOGC) + c4);
  float* o = agg + (dst << LOGC) + c4;
  atomicAdd(o + 0, v.x * nrm);
  atomicAdd(o + 1, v.y * nrm);
  atomicAdd(o + 2, v.z * nrm);
  atomicAdd(o + 3, v.w * nrm);
}

// single-channel variant for the final layer
__global__ void scatter1_kernel(const float* __restrict__ ht, const int4* __restrict__ edges,
                                float* __restrict__ agg, unsigned E) {
  unsigned e = blockIdx.x * blockDim.x + threadIdx.x;
  if (e >= E) return;
  const int4 r = edges[e];
  atomicAdd(&agg[(unsigned)r.y], ht[(unsigned)r.x] * __int_as_float(r.z));
}

// out = (agg + dis[i]^2 * ht + bias[c]) with optional tanh; in-place capable.
template <int LOGC, bool TANH>
__global__ void finalize_kernel(const float* __restrict__ agg, const float* __restrict__ ht,
                                const float* __restrict__ dis, const float* __restrict__ bias,
                                float* __restrict__ out, unsigned total) {
  unsigned t = blockIdx.x * blockDim.x + threadIdx.x;
  if (t >= total) return;
  unsigned i = t >> LOGC;
  unsigned c = t & ((1u << LOGC) - 1u);
  float d = dis[i];
  float v = agg[t] + d * d * ht[t] + bias[c];
  out[t] = TANH ? tanhf(v) : v;
}

// out[i] = H[i,0:32] . W3   (layer-3 dense, OUT=1: plain VALU dot)
__global__ void dot32_kernel(const float* __restrict__ H, const float* __restrict__ W3,
                             float* __restrict__ out, int N) {
  int i = blockIdx.x * blockDim.x + threadIdx.x;
  if (i >= N) return;
  const float4* h = (const float4*)(H + (unsigned)i * 32u);
  float s = 0.f;
#pragma unroll
  for (int q = 0; q < 8; ++q) {
    float4 a = h[q];
    float4 w = ((const float4*)W3)[q];
    s += a.x * w.x + a.y * w.y + a.z * w.z + a.w * w.w;
  }
  out[i] = s;
}

extern "C" void kernel_launch(void* const* d_in, const int* in_sizes, int n_in,
                              void* d_out, int out_size, void* d_ws, size_t ws_size,
                              hipStream_t stream) {
  const float* x  = (const float*)d_in[0];
  const int*   ei = (const int*)d_in[1];   // [2,E]: row 0 = src, row 1 = dst
  const float* W1 = (const float*)d_in[2];
  const float* b1 = (const float*)d_in[3];
  const float* W2 = (const float*)d_in[4];
  const float* b2 = (const float*)d_in[5];
  const float* W3 = (const float*)d_in[6];
  const float* b3 = (const float*)d_in[7];
  float* out = (float*)d_out;

  const int D2 = 64, D = 32;
  const int N = in_sizes[0] / D2;
  const int E = in_sizes[1] / 2;

  // Workspace layout (floats), ~78 MB total, L2-resident working set.
  float* ws    = (float*)d_ws;
  float* dis   = ws;                              // N
  float* ht    = ws + N;                          // N*64 (transformed features)
  float* agg   = ht + (long long)N * D2;          // N*64 (aggregate; h1/h2 in-place)
  float* ht3   = agg + (long long)N * D2;         // N
  float* agg3  = ht3 + N;                         // N
  int4*  edges = (int4*)(agg3 + N);               // E * 16B packed records
  float* deg   = ht;                              // alias: dead before gemm writes ht
  (void)ws_size; (void)n_in; (void)out_size;

  const int B = 256;  // 8 wave32s per block

  // --- degrees, symmetric norm, packed edge records ---
  fill_kernel<<<(unsigned)cdiv_ll(N, B), B, 0, stream>>>(deg, 1.0f, (unsigned)N);
  deg_kernel<<<(unsigned)cdiv_ll(E, B), B, 0, stream>>>(ei, deg, E);
  dis_kernel<<<(unsigned)cdiv_ll(N, B), B, 0, stream>>>(deg, dis, N);
  edge_kernel<<<(unsigned)cdiv_ll(E, B), B, 0, stream>>>(ei, dis, edges, E);

  const int rowTiles = (int)cdiv_ll(N, 16);

  // --- layer 1: 64 -> 64, tanh ---
  {
    int colT = D2 / 16, tiles = rowTiles * colT;
    gemm_wmma_f32<<<(unsigned)cdiv_ll((long long)tiles * 32, B), B, 0, stream>>>(
        x, W1, ht, N, D2, D2, colT, tiles);
  }
  fill_kernel<<<(unsigned)cdiv_ll((long long)N * D2, B), B, 0, stream>>>(agg, 0.f, (unsigned)(N * D2));
  scatter_kernel<6><<<(unsigned)cdiv_ll((long long)E * (D2 / 4), B), B, 0, stream>>>(ht, edges, agg, (unsigned)E);
  finalize_kernel<6, true><<<(unsigned)cdiv_ll((long long)N * D2, B), B, 0, stream>>>(
      agg, ht, dis, b1, agg, (unsigned)(N * D2));

  // --- layer 2: 64 -> 32, tanh (h1 lives in `agg`, read before re-zeroing) ---
  {
    int colT = D / 16, tiles = rowTiles * colT;
    gemm_wmma_f32<<<(unsigned)cdiv_ll((long long)tiles * 32, B), B, 0, stream>>>(
        agg, W2, ht, N, D2, D, colT, tiles);
  }
  fill_kernel<<<(unsigned)cdiv_ll((long long)N * D, B), B, 0, stream>>>(agg, 0.f, (unsigned)(N * D));
  scatter_kernel<5><<<(unsigned)cdiv_ll((long long)E * (D / 4), B), B, 0, stream>>>(ht, edges, agg, (unsigned)E);
  finalize_kernel<5, true><<<(unsigned)cdiv_ll((long long)N * D, B), B, 0, stream>>>(
      agg, ht, dis, b2, agg, (unsigned)(N * D));

  // --- layer 3: 32 -> 1, no tanh ---
  dot32_kernel<<<(unsigned)cdiv_ll(N, B), B, 0, stream>>>(agg, W3, ht3, N);
  fill_kernel<<<(unsigned)cdiv_ll(N, B), B, 0, stream>>>(agg3, 0.f, (unsigned)N);
  scatter1_kernel<<<(unsigned)cdiv_ll(E, B), B, 0, stream>>>(ht3, edges, agg3, (unsigned)E);
  finalize_kernel<0, false><<<(unsigned)cdiv_ll(N, B), B, 0, stream>>>(
      agg3, ht3, dis, b3, out, (unsigned)N);
}